// GNN_25804163515003
// MI455X (gfx1250) — compile-verified
//
#include <hip/hip_runtime.h>

#define N_NODES 100000
#define N_EDGES 1600000
#define FEAT 64
#define OUT_CH 16

typedef __attribute__((ext_vector_type(2))) float v2f;
typedef __attribute__((ext_vector_type(8))) float v8f;

// ---------------------------------------------------------------------------
// Degree count: one thread per edge.
// ---------------------------------------------------------------------------
__global__ __launch_bounds__(256) void gnn_count_kernel(
    const long long* __restrict__ ei, float* __restrict__ cnt)
{
    int e = blockIdx.x * 256 + threadIdx.x;
    long long dst = ei[N_EDGES + e];
    atomicAdd(&cnt[dst], 1.0f);
}

// ---------------------------------------------------------------------------
// Scatter-add messages: 16 threads per edge, float4 per thread (64 ch total).
// Consecutive lanes read consecutive 16B chunks of the same source row
// (coalesced 256B per edge), then 4 float atomics into the dst row.
// ---------------------------------------------------------------------------
__global__ __launch_bounds__(256) void gnn_scatter_kernel(
    const long long* __restrict__ ei, const float* __restrict__ x,
    float* __restrict__ agg)
{
    int t = blockIdx.x * 256 + threadIdx.x;
    int e = t >> 4;
    int chunk = t & 15;
    long long src = ei[e];
    long long dst = ei[N_EDGES + e];
    float4 v = *((const float4*)(x + src * FEAT) + chunk);
    float* o = agg + dst * FEAT + chunk * 4;
    atomicAdd(o + 0, v.x);
    atomicAdd(o + 1, v.y);
    atomicAdd(o + 2, v.z);
    atomicAdd(o + 3, v.w);
}

// ---------------------------------------------------------------------------
// Fused SAGEConv: out = relu( (agg/deg) @ Wl^T + b + x @ Wr^T )
// 128 threads = 4 waves per 16-node tile; wave w computes out cols [16w,16w+16).
// fp32 WMMA 16x16x4, K=64 in 16 steps, 2 wmma per step (agg term + self term).
//
// A frag (16x4 f32): lanes 0-15 hold K=k0..k0+1 of row=lane; lanes 16-31 hold
// K=k0+2..k0+3  -> one float2 load at row*64 + k0 + (lane>>4)*2.
// B frag (4x16, B[k][o] = W[o*64+k]): identical float2 pattern on W rows.
// ---------------------------------------------------------------------------
__global__ __launch_bounds__(128) void gnn_sage_kernel(
    const float* __restrict__ xin, const float* __restrict__ agg,
    const float* __restrict__ cnt,
    const float* __restrict__ Wl, const float* __restrict__ bvec,
    const float* __restrict__ Wr, float* __restrict__ out)
{
    const int tile  = blockIdx.x;
    const int lane  = threadIdx.x & 31;
    const int wave  = threadIdx.x >> 5;
    const int row   = lane & 15;
    const int khalf = lane >> 4;
    const int obase = wave * 16;

    const int node = tile * 16 + row;
    const float scale = 1.0f / fmaxf(cnt[node], 1.0f);

    const float* Arow  = agg + (size_t)node * FEAT + khalf * 2;
    const float* Xrow  = xin + (size_t)node * FEAT + khalf * 2;
    const float* WlRow = Wl + (size_t)(obase + row) * FEAT + khalf * 2;
    const float* WrRow = Wr + (size_t)(obase + row) * FEAT + khalf * 2;

    v8f acc = {0.f, 0.f, 0.f, 0.f, 0.f, 0.f, 0.f, 0.f};
#pragma unroll
    for (int k0 = 0; k0 < FEAT; k0 += 4) {
        v2f a = *(const v2f*)(Arow + k0);
        a *= scale;
        v2f bw = *(const v2f*)(WlRow + k0);
        acc = __builtin_amdgcn_wmma_f32_16x16x4_f32(
            false, a, false, bw, (short)0, acc, false, false);
        v2f a2  = *(const v2f*)(Xrow + k0);
        v2f bw2 = *(const v2f*)(WrRow + k0);
        acc = __builtin_amdgcn_wmma_f32_16x16x4_f32(
            false, a2, false, bw2, (short)0, acc, false, false);
    }

    const float bias = bvec[obase + row];
#pragma unroll
    for (int r = 0; r < 8; ++r) {
        int m = r + khalf * 8;                   // C/D layout: VGPR r -> M=r / r+8
        float v = fmaxf(acc[r] + bias, 0.0f);
        out[(size_t)(tile * 16 + m) * FEAT + obase + row] = v;
    }
}

// ---------------------------------------------------------------------------
// Output head: out[N,16] = h @ Wlin^T + blin. One wave per 16-node tile.
// ---------------------------------------------------------------------------
__global__ __launch_bounds__(64) void gnn_out_kernel(
    const float* __restrict__ h, const float* __restrict__ Wlin,
    const float* __restrict__ blin, float* __restrict__ out)
{
    const int lane  = threadIdx.x & 31;
    const int wave  = threadIdx.x >> 5;
    const int tile  = blockIdx.x * 2 + wave;
    const int row   = lane & 15;
    const int khalf = lane >> 4;

    const float* Hrow = h + (size_t)(tile * 16 + row) * FEAT + khalf * 2;
    const float* Wrow = Wlin + (size_t)row * FEAT + khalf * 2;

    v8f acc = {0.f, 0.f, 0.f, 0.f, 0.f, 0.f, 0.f, 0.f};
#pragma unroll
    for (int k0 = 0; k0 < FEAT; k0 += 4) {
        v2f a  = *(const v2f*)(Hrow + k0);
        v2f bw = *(const v2f*)(Wrow + k0);
        acc = __builtin_amdgcn_wmma_f32_16x16x4_f32(
            false, a, false, bw, (short)0, acc, false, false);
    }

    const float bias = blin[row];
#pragma unroll
    for (int r = 0; r < 8; ++r) {
        int m = r + khalf * 8;
        out[(size_t)(tile * 16 + m) * OUT_CH + row] = acc[r] + bias;
    }
}

// ---------------------------------------------------------------------------
extern "C" void kernel_launch(void* const* d_in, const int* in_sizes, int n_in,
                              void* d_out, int out_size, void* d_ws, size_t ws_size,
                              hipStream_t stream) {
    const float*     x    = (const float*)d_in[0];
    const long long* ei   = (const long long*)d_in[1];
    const float*     W1l  = (const float*)d_in[2];
    const float*     b1   = (const float*)d_in[3];
    const float*     W1r  = (const float*)d_in[4];
    const float*     W2l  = (const float*)d_in[5];
    const float*     b2   = (const float*)d_in[6];
    const float*     W2r  = (const float*)d_in[7];
    const float*     Wlin = (const float*)d_in[8];
    const float*     blin = (const float*)d_in[9];
    float* out = (float*)d_out;

    char* ws = (char*)d_ws;
    // 256B-aligned workspace layout (total ~77.2 MB)
    float* cnt = (float*)(ws + 0);                     //  400,000 B
    float* agg = (float*)(ws + 400128);                // 25.6 MB
    float* h1  = (float*)(ws + 26000128);              // 25.6 MB
    float* h2  = (float*)(ws + 51600128);              // 25.6 MB

    const int TILES = N_NODES / 16;                    // 6250 exactly

    // ---- layer 1 ----
    hipMemsetAsync(cnt, 0, (size_t)N_NODES * sizeof(float), stream);
    hipMemsetAsync(agg, 0, (size_t)N_NODES * FEAT * sizeof(float), stream);
    gnn_count_kernel<<<N_EDGES / 256, 256, 0, stream>>>(ei, cnt);
    gnn_scatter_kernel<<<(N_EDGES * 16) / 256, 256, 0, stream>>>(ei, x, agg);
    gnn_sage_kernel<<<TILES, 128, 0, stream>>>(x, agg, cnt, W1l, b1, W1r, h1);

    // ---- layer 2 (degree counts reused) ----
    hipMemsetAsync(agg, 0, (size_t)N_NODES * FEAT * sizeof(float), stream);
    gnn_scatter_kernel<<<(N_EDGES * 16) / 256, 256, 0, stream>>>(ei, h1, agg);
    gnn_sage_kernel<<<TILES, 128, 0, stream>>>(h1, agg, cnt, W2l, b2, W2r, h2);

    // ---- output head ----
    gnn_out_kernel<<<TILES / 2, 64, 0, stream>>>(h2, Wlin, blin, out);
}